// FeatureAggregation_17051020165783
// MI455X (gfx1250) — compile-verified
//
#include <hip/hip_runtime.h>
#include <math.h>

typedef __attribute__((ext_vector_type(2))) float v2f;
typedef __attribute__((ext_vector_type(8))) float v8f;

typedef __attribute__((address_space(1))) int as1_int;
typedef __attribute__((address_space(3))) int as3_int;

#define NPTS  2048
#define BATCH 8
#define CH    64
#define KNB   32
#define EPSV  1e-5f
#define BIGF  3.0e38f

#if defined(__has_builtin)
#  if __has_builtin(__builtin_amdgcn_global_load_async_to_lds_b32)
#    define HAVE_ASYNC_LDS 1
#  endif
#endif
#ifndef HAVE_ASYNC_LDS
#  define HAVE_ASYNC_LDS 0
#endif

// ---------------------------------------------------------------------------
// ws layout (4-byte units):
//   float acc[16]   : (sum, sumsq) per batch
//   float stdv[8]   : per-batch std
//   int   idx[B*N*K]: neighbor indices (starts at float offset 32)
// ---------------------------------------------------------------------------

__global__ void zero_acc_kernel(float* acc) {
    if (threadIdx.x < 2 * BATCH) acc[threadIdx.x] = 0.0f;
}

// ---------------------------------------------------------------------------
// Kernel 1: distance slab via V_WMMA_F32_16X16X4_F32 + in-LDS top-32 select.
// grid = B * (N/16), block = 256 (8 wave32), dynamic LDS = 160 KB.
// ---------------------------------------------------------------------------
__global__ __launch_bounds__(256)
void knn_topk_kernel(const float* __restrict__ xyz, int* __restrict__ idxOut) {
    extern __shared__ float smem[];
    float* sx   = smem;               // [2048]
    float* sy   = sx + NPTS;          // [2048]
    float* sz   = sy + NPTS;          // [2048]
    float* ssq  = sz + NPTS;          // [2048]
    float* dist = ssq + NPTS;         // [16][2048] (also reused as raw stage)

    const int b     = blockIdx.x / (NPTS / 16);
    const int rbase = (blockIdx.x % (NPTS / 16)) * 16;
    const int tid   = threadIdx.x;
    const int lane  = tid & 31;
    const int wave  = tid >> 5;

    const float* bx = xyz + (size_t)b * NPTS * 3;

#if HAVE_ASYNC_LDS
    // --- Stage 0a: async copy raw xyz block (2048*3 dwords) into LDS -------
    // GLOBAL_LOAD_ASYNC_TO_LDS_B32: per-lane global addr -> per-lane LDS addr,
    // tracked on ASYNCcnt. Lands in the dist region (overwritten later).
    {
        as1_int* gsrc = (as1_int*)(uintptr_t)(const void*)bx;
        as3_int* lraw = (as3_int*)(uint32_t)(uintptr_t)(void*)dist;
        for (int j = tid; j < NPTS * 3; j += 256) {
            __builtin_amdgcn_global_load_async_to_lds_b32(gsrc + j, lraw + j, 0, 0);
        }
#  if __has_builtin(__builtin_amdgcn_s_wait_asynccnt)
        __builtin_amdgcn_s_wait_asynccnt(0);
#  else
        asm volatile("s_wait_asynccnt 0" ::: "memory");
#  endif
    }
    __syncthreads();
    // --- Stage 0b: build SoA x/y/z/|p|^2 from LDS raw -----------------------
    for (int i = tid; i < NPTS; i += 256) {
        float x = dist[i * 3 + 0];
        float y = dist[i * 3 + 1];
        float z = dist[i * 3 + 2];
        sx[i] = x; sy[i] = y; sz[i] = z;
        ssq[i] = x * x + y * y + z * z;
    }
#else
    // Fallback: direct global staging.
    for (int i = tid; i < NPTS; i += 256) {
        float x = bx[i * 3 + 0];
        float y = bx[i * 3 + 1];
        float z = bx[i * 3 + 2];
        sx[i] = x; sy[i] = y; sz[i] = z;
        ssq[i] = x * x + y * y + z * z;
    }
#endif
    __syncthreads();

    // --- Phase 1: 16 x 2048 distance slab, WMMA f32 16x16x4 ----------------
    // A layout (16x4 f32): v0 = lanes<16 -> K=0 (x), lanes>=16 -> K=2 (z)
    //                      v1 = lanes<16 -> K=1 (y), lanes>=16 -> K=3 (pad 0)
    const int  l  = lane & 15;
    const bool hi = (lane >= 16);

    // Unconditional loads + register selects (avoid exec-masked loads).
    {
        const float rx = sx[rbase + l];
        const float ry = sy[rbase + l];
        const float rz = sz[rbase + l];
        v2f afrag;
        afrag.x = hi ? rz : rx;
        afrag.y = hi ? 0.0f : ry;

        for (int t = 0; t < 16; ++t) {            // each wave: 16 column tiles
            const int cb = (wave + t * 8) * 16;   // column base
            const float cx = sx[cb + l];
            const float cy = sy[cb + l];
            const float cz = sz[cb + l];
            v2f bfrag;                            // B (4x16) mirrors A's striping
            bfrag.x = hi ? cz : cx;
            bfrag.y = hi ? 0.0f : cy;

            v8f acc = {};
            acc = __builtin_amdgcn_wmma_f32_16x16x4_f32(
                      false, afrag, false, bfrag, (short)0, acc, false, false);

            // D layout: VGPR i -> M = i + 8*hi, N = lane&15
            const int mb = hi ? 8 : 0;
            #pragma unroll
            for (int i = 0; i < 8; ++i) {
                const int m = mb + i;
                dist[m * NPTS + cb + l] =
                    ssq[rbase + m] + ssq[cb + l] - 2.0f * acc[i];
            }
        }
    }
    __syncthreads();

    // --- Phase 2: top-32 smallest per row (wave handles 2 rows) ------------
    for (int rr = 0; rr < 2; ++rr) {
        const int m = wave * 2 + rr;
        float* drow = dist + m * NPTS;

        // cached per-lane argmin over strided chunk {lane, lane+32, ...}
        float lmin = BIGF; int lidx = lane;
        for (int t = 0; t < NPTS / 32; ++t) {
            const int j = lane + (t << 5);
            const float v = drow[j];
            if (v < lmin) { lmin = v; lidx = j; }
        }

        int* rowOut = idxOut + ((size_t)(b * NPTS + rbase + m)) * KNB;
        for (int sel = 0; sel < KNB; ++sel) {
            float wmin = lmin; int widx = lidx;
            #pragma unroll
            for (int off = 16; off > 0; off >>= 1) {
                const float ov = __shfl_xor(wmin, off, 32);
                const int   oi = __shfl_xor(widx, off, 32);
                if (ov < wmin || (ov == wmin && oi < widx)) { wmin = ov; widx = oi; }
            }
            if (lane == 0) rowOut[sel] = widx;
            if ((widx & 31) == lane) {        // winner lane: remove + rescan
                drow[widx] = BIGF;
                lmin = BIGF; lidx = lane;
                for (int t = 0; t < NPTS / 32; ++t) {
                    const int j = lane + (t << 5);
                    const float v = drow[j];
                    if (v < lmin) { lmin = v; lidx = j; }
                }
            }
        }
    }
}

// ---------------------------------------------------------------------------
// Kernel 2: per-batch sum / sumsq of diff = grouped - anchor.
// grid = B*N, block = 256; float4 gathers, LDS tree reduce, 2 atomics/block.
// ---------------------------------------------------------------------------
__global__ __launch_bounds__(256)
void moments_kernel(const float* __restrict__ points, const int* __restrict__ idx,
                    float* __restrict__ acc) {
    __shared__ float4 anchor[CH / 4];
    __shared__ int    nb[KNB];
    __shared__ float  rs[256], rs2[256];

    const int b   = blockIdx.x / NPTS;
    const int n   = blockIdx.x % NPTS;
    const int tid = threadIdx.x;

    const float4* p4 = (const float4*)points;
    const size_t rowBase = ((size_t)b * NPTS + n) * (CH / 4);
    if (tid < CH / 4) anchor[tid] = p4[rowBase + tid];
    if (tid < KNB)    nb[tid] = idx[((size_t)b * NPTS + n) * KNB + tid];
    __syncthreads();

    if (tid < KNB) {   // prefetch gather rows -> global_prefetch_b8
        __builtin_prefetch(&points[((size_t)b * NPTS + nb[tid]) * CH], 0, 0);
    }

    float s = 0.0f, s2 = 0.0f;
    #pragma unroll
    for (int i = 0; i < 2; ++i) {
        const int e  = tid + i * 256;   // 0..511 float4 slots (32 k x 16 c4)
        const int k  = e >> 4;
        const int c4 = e & 15;
        const float4 q = p4[((size_t)b * NPTS + nb[k]) * (CH / 4) + c4];
        const float4 a = anchor[c4];
        const float dx = q.x - a.x, dy = q.y - a.y, dz = q.z - a.z, dw = q.w - a.w;
        s  += dx + dy + dz + dw;
        s2 += dx * dx + dy * dy + dz * dz + dw * dw;
    }
    rs[tid] = s; rs2[tid] = s2;
    __syncthreads();
    for (int st = 128; st > 0; st >>= 1) {
        if (tid < st) { rs[tid] += rs[tid + st]; rs2[tid] += rs2[tid + st]; }
        __syncthreads();
    }
    if (tid == 0) {
        atomicAdd(&acc[b * 2 + 0], rs[0]);
        atomicAdd(&acc[b * 2 + 1], rs2[0]);
    }
}

__global__ void finalize_std_kernel(const float* __restrict__ acc,
                                    float* __restrict__ stdv) {
    const int t = threadIdx.x;
    if (t < BATCH) {
        const float M  = (float)((size_t)NPTS * KNB * CH);
        const float s  = acc[t * 2 + 0];
        const float s2 = acc[t * 2 + 1];
        const float mean = s / M;
        const float var  = (s2 - s * mean) / (M - 1.0f);
        stdv[t] = sqrtf(fmaxf(var, 0.0f));
    }
}

// ---------------------------------------------------------------------------
// Kernel 3: streaming writer. grid = B*N, block = 256, all-float4 stores.
// out row per k = [ g(64 f32) | anchor(64 f32) ] = 32 float4.
// ---------------------------------------------------------------------------
__global__ __launch_bounds__(256)
void writer_kernel(const float* __restrict__ points, const int* __restrict__ idx,
                   const float* __restrict__ alpha, const float* __restrict__ beta,
                   const float* __restrict__ stdv, float* __restrict__ outNP) {
    __shared__ float4 anchor[CH / 4];
    __shared__ float4 sal[CH / 4], sbe[CH / 4];
    __shared__ int    nb[KNB];

    const int b   = blockIdx.x / NPTS;
    const int n   = blockIdx.x % NPTS;
    const int tid = threadIdx.x;

    const float4* p4  = (const float4*)points;
    const float4* al4 = (const float4*)alpha;
    const float4* be4 = (const float4*)beta;
    if (tid < CH / 4) {
        anchor[tid] = p4[((size_t)b * NPTS + n) * (CH / 4) + tid];
        sal[tid] = al4[tid];
        sbe[tid] = be4[tid];
    }
    if (tid < KNB) nb[tid] = idx[((size_t)b * NPTS + n) * KNB + tid];
    __syncthreads();

    if (tid < KNB) {   // prefetch gather rows -> global_prefetch_b8
        __builtin_prefetch(&points[((size_t)b * NPTS + nb[tid]) * CH], 0, 0);
    }

    const float inv = 1.0f / (stdv[b] + EPSV);
    float4* out4 = (float4*)outNP + ((size_t)b * NPTS + n) * (KNB * 32);

    #pragma unroll
    for (int i = 0; i < 4; ++i) {
        const int q    = tid + i * 256;   // 0..1023: k = q/32, slot = q%32
        const int k    = q >> 5;
        const int slot = q & 31;
        float4 r;
        if (slot < 16) {                  // g half
            const float4 g = p4[((size_t)b * NPTS + nb[k]) * (CH / 4) + slot];
            const float4 a = anchor[slot];
            const float4 av = sal[slot], bv = sbe[slot];
            r.x = av.x * ((g.x - a.x) * inv) + bv.x;
            r.y = av.y * ((g.y - a.y) * inv) + bv.y;
            r.z = av.z * ((g.z - a.z) * inv) + bv.z;
            r.w = av.w * ((g.w - a.w) * inv) + bv.w;
        } else {                          // anchor half
            r = anchor[slot - 16];
        }
        out4[(size_t)k * 32 + slot] = r;
    }
}

// ---------------------------------------------------------------------------
extern "C" void kernel_launch(void* const* d_in, const int* in_sizes, int n_in,
                              void* d_out, int out_size, void* d_ws, size_t ws_size,
                              hipStream_t stream) {
    const float* xyz    = (const float*)d_in[0];
    const float* points = (const float*)d_in[1];
    const float* alpha  = (const float*)d_in[2];
    const float* beta   = (const float*)d_in[3];

    float* out  = (float*)d_out;
    float* acc  = (float*)d_ws;          // 16 floats
    float* stdv = acc + 16;              // 8 floats
    int*   idx  = (int*)d_ws + 32;       // B*N*K ints

    // output head: xyz passthrough (8*2048*3 f32)
    (void)hipMemcpyAsync(out, xyz, (size_t)BATCH * NPTS * 3 * sizeof(float),
                         hipMemcpyDeviceToDevice, stream);

    zero_acc_kernel<<<1, 32, 0, stream>>>(acc);

    const size_t knnLds = (size_t)(4 * NPTS + 16 * NPTS) * sizeof(float); // 160 KB
    knn_topk_kernel<<<BATCH * (NPTS / 16), 256, knnLds, stream>>>(xyz, idx);

    moments_kernel<<<BATCH * NPTS, 256, 0, stream>>>(points, idx, acc);
    finalize_std_kernel<<<1, 32, 0, stream>>>(acc, stdv);

    writer_kernel<<<BATCH * NPTS, 256, 0, stream>>>(
        points, idx, alpha, beta, stdv, out + (size_t)BATCH * NPTS * 3);
}